// LocalDynamicGraph_79594333929751
// MI455X (gfx1250) — compile-verified
//
#include <hip/hip_runtime.h>

typedef float v4f __attribute__((ext_vector_type(4)));

#define HAS_ASYNC_LDS __has_builtin(__builtin_amdgcn_global_load_async_to_lds_b32)

// Builtin pointee/AS types: global (AS1) and LDS (AS3) int, per hipcc diagnostic.
typedef __attribute__((address_space(1))) int gint_t;
typedef __attribute__((address_space(3))) int lint_t;

constexpr int B = 8, N = 8192, C = 64, K = 16;
constexpr int PT_OUT = 2 * C * K;   // 2048 floats of output per point

// One block = one (b, n) point.
//  - gather 16 neighbor rows (64 floats each) TRANSPOSED into LDS featT[c][k]
//    via CDNA5 async global->LDS copies (ASYNCcnt), falling back to ds stores
//  - lane t then reads LDS linearly (float4 @ byte 16*t), subtracts its center
//    channel, and writes two contiguous non-temporal float4s:
//      f4 = t       -> out[c = t/4      ][k = 4*(t%4)..+3] = feat - center
//      f4 = t + 256 -> out[c = 64 + t/4 ][k..] = center broadcast
__global__ __launch_bounds__(256) void LocalDynamicGraph_edge_kernel(
    const float* __restrict__ x, const int* __restrict__ idx,
    float* __restrict__ out)
{
    __shared__ __align__(16) float featT[C * K];   // [c][k], 4 KB

    const int t  = threadIdx.x;        // 0..255
    const int bn = blockIdx.x;         // b*N + n
    const int b  = bn >> 13;           // bn / N

    const float* xb = x   + (size_t)b  * N * C;    // batch base of x
    const int*   ib = idx + (size_t)bn * K;        // 16 neighbor indices

    // Center value this lane needs: channel c = t/4 of point (b, n).
    const float cen = x[(size_t)bn * C + (t >> 2)];

    // Gather phase: 1024 elements, 4 per lane. e -> (k = e/64, c = e%64),
    // scattered transposed into LDS at dword (c*16 + k).
#pragma unroll
    for (int i = 0; i < 4; ++i) {
        const int e   = t + i * 256;
        const int k   = e >> 6;
        const int c   = e & 63;
        const int row = ib[k];
        const float* gp = xb + (size_t)row * C + c;
        float*       lp = &featT[c * K + k];
#if HAS_ASYNC_LDS
        __builtin_amdgcn_global_load_async_to_lds_b32(
            (gint_t*)gp, (lint_t*)lp, /*offset=*/0, /*cpol=*/0);
#else
        *lp = *gp;                       // plain load + ds_store fallback
#endif
    }

#if HAS_ASYNC_LDS
#if __has_builtin(__builtin_amdgcn_s_wait_asynccnt)
    __builtin_amdgcn_s_wait_asynccnt(0);
#else
    asm volatile("s_wait_asynccnt 0" ::: "memory");
#endif
#endif
    __syncthreads();

    // Write phase: 8 KB contiguous output for this point.
    float* op = out + (size_t)bn * PT_OUT;

    const v4f f4 = ((const v4f*)featT)[t];          // linear, conflict-free
    v4f o0;
    o0.x = f4.x - cen; o0.y = f4.y - cen; o0.z = f4.z - cen; o0.w = f4.w - cen;
    v4f o1;
    o1.x = cen; o1.y = cen; o1.z = cen; o1.w = cen;

    // Non-temporal: keep the 512 MB output stream from evicting L2-resident x.
    __builtin_nontemporal_store(o0, ((v4f*)op) + t);
    __builtin_nontemporal_store(o1, ((v4f*)op) + t + 256);
}

extern "C" void kernel_launch(void* const* d_in, const int* in_sizes, int n_in,
                              void* d_out, int out_size, void* d_ws, size_t ws_size,
                              hipStream_t stream) {
    const float* x   = (const float*)d_in[0];
    const int*   idx = (const int*)d_in[1];   // JAX w/o x64 -> int32 indices
    float*       out = (float*)d_out;

    dim3 grid(B * N);   // 65536 blocks, one per (b, n)
    dim3 block(256);    // 8 waves (wave32)
    LocalDynamicGraph_edge_kernel<<<grid, block, 0, stream>>>(x, idx, out);
}